// gu_HiPPO_LTI_46574625358288
// MI455X (gfx1250) — compile-verified
//
#include <hip/hip_runtime.h>
#include <cstddef>

typedef __attribute__((ext_vector_type(2))) float v2f;
typedef __attribute__((ext_vector_type(8))) float v8f;

#define Nn 256          // state dim
#define Ll 2048         // sequence length
#define Bb 128          // batch
#define CHUNK 64        // timesteps per parallel chunk
#define NCHUNK (Ll / CHUNK)   // 32
#define BROWS 16        // batch rows per workgroup (one WMMA M-tile)
#define NBB (Bb / BROWS)      // 8 batch blocks
#define XSTR 258        // padded LDS row stride (floats): 258 % 64 == 2 -> no bank clash

__device__ __forceinline__ v8f wmma4(v2f a, v2f b, v8f c) {
  // D = A(16x4 f32) * B(4x16 f32) + C(16x16 f32)
  return __builtin_amdgcn_wmma_f32_16x16x4_f32(false, a, false, b, (short)0, c,
                                               false, false);
}

__device__ __forceinline__ v8f zero8() {
  v8f z;
#pragma unroll
  for (int v = 0; v < 8; ++v) z[v] = 0.0f;
  return z;
}

// ---------------------------------------------------------------- phase 0 ---
// G = dA^T  (row-vector convention: x_new = x @ G)
__global__ void k_transpose(const float* __restrict__ dA, float* __restrict__ G) {
  int i = blockIdx.x * blockDim.x + threadIdx.x;  // 0..65535
  int r = i >> 8, c = i & 255;
  G[r * Nn + c] = dA[c * Nn + r];
}

// C = A @ B, all 256x256 row-major. 16 blocks x 16 waves = 256 output tiles.
__global__ __launch_bounds__(512, 1)
void k_matmul256(const float* __restrict__ A, const float* __restrict__ B,
                 float* __restrict__ C) {
  const int tid = threadIdx.x, lane = tid & 31, wave = tid >> 5;
  const int lo = lane & 15, hi = lane >> 4;
  const int tile = blockIdx.x * 16 + wave;
  const int m0 = (tile >> 4) * 16, n0 = (tile & 15) * 16;
  v8f c0 = zero8(), c1 = zero8(), c2 = zero8(), c3 = zero8();
#pragma unroll
  for (int kb = 0; kb < 64; kb += 4) {
#pragma unroll
    for (int q = 0; q < 4; ++q) {
      const int k0 = 4 * (kb + q) + 2 * hi;
      v2f a, b;
      const float* ap = A + (m0 + lo) * Nn + k0;
      a.x = ap[0];
      a.y = ap[1];
      b.x = B[(size_t)k0 * Nn + n0 + lo];
      b.y = B[(size_t)(k0 + 1) * Nn + n0 + lo];
      if (q == 0) c0 = wmma4(a, b, c0);
      else if (q == 1) c1 = wmma4(a, b, c1);
      else if (q == 2) c2 = wmma4(a, b, c2);
      else c3 = wmma4(a, b, c3);
    }
  }
  v8f c = (c0 + c1) + (c2 + c3);
#pragma unroll
  for (int v = 0; v < 8; ++v)
    C[(size_t)(m0 + v + 8 * hi) * Nn + n0 + lo] = c[v];
}

// ---------------------------------------------------------------- phase 1 ---
// Local scan of one 64-step chunk with x0 = 0; writes out[t] for t in chunk.
__global__ __launch_bounds__(512, 1)
void k_scan_local(const float* __restrict__ inp, const float* __restrict__ dA,
                  const float* __restrict__ dB, float* __restrict__ out) {
  __shared__ float xb[2][BROWS * XSTR];
  const int tid = threadIdx.x, lane = tid & 31, wave = tid >> 5;
  const int lo = lane & 15, hi = lane >> 4;
  const int n0 = wave * 16, nc = n0 + lo;
  const int b0 = blockIdx.x * BROWS;
  const int t0 = blockIdx.y * CHUNK;

  // Resident B-fragments of dA^T: B[j][n] = dA[n][j]  (64 x float2 = 128 VGPRs)
  v2f bfr[64];
#pragma unroll
  for (int kb = 0; kb < 64; ++kb) {
    const float* p = dA + (size_t)nc * Nn + 4 * kb + 2 * hi;
    bfr[kb].x = p[0];
    bfr[kb].y = p[1];
  }
  const float dbv = dB[nc];

  for (int i = tid; i < BROWS * XSTR; i += 512) xb[0][i] = 0.0f;
  __syncthreads();

  int cur = 0;
#pragma unroll 1
  for (int j = 0; j < CHUNK; ++j) {
    const int t = t0 + j;
    const float* it = inp + (size_t)t * Bb + b0 + 8 * hi;
    v8f c0, c1 = zero8(), c2 = zero8(), c3 = zero8();
#pragma unroll
    for (int v = 0; v < 8; ++v) c0[v] = it[v] * dbv;  // C init = u_t tile

    const float* xc = &xb[cur][lo * XSTR + 2 * hi];
#pragma unroll
    for (int kb = 0; kb < 64; kb += 4) {
      v2f a0 = *(const v2f*)(xc + 4 * kb);
      v2f a1 = *(const v2f*)(xc + 4 * kb + 4);
      v2f a2 = *(const v2f*)(xc + 4 * kb + 8);
      v2f a3 = *(const v2f*)(xc + 4 * kb + 12);
      c0 = wmma4(a0, bfr[kb + 0], c0);
      c1 = wmma4(a1, bfr[kb + 1], c1);
      c2 = wmma4(a2, bfr[kb + 2], c2);
      c3 = wmma4(a3, bfr[kb + 3], c3);
    }
    v8f c = (c0 + c1) + (c2 + c3);

    float* xn = xb[cur ^ 1];
    float* og = out + ((size_t)t * Bb + b0 + 8 * hi) * Nn + nc;
#pragma unroll
    for (int v = 0; v < 8; ++v) {
      xn[(v + 8 * hi) * XSTR + nc] = c[v];
      og[(size_t)v * Nn] = c[v];
    }
    __syncthreads();
    cur ^= 1;
  }
}

// ---------------------------------------------------------------- phase 2 ---
// Sequential over chunk boundaries: B_0 = S_0; B_k = S_k + B_{k-1} @ M.
// S_k = out[k*CHUNK + CHUNK - 1] (local final state). Stores B_k to ws.
__global__ __launch_bounds__(512, 1)
void k_boundary(const float* __restrict__ out, const float* __restrict__ M,
                float* __restrict__ Bws) {
  __shared__ float xb[2][BROWS * XSTR];
  const int tid = threadIdx.x, lane = tid & 31, wave = tid >> 5;
  const int lo = lane & 15, hi = lane >> 4;
  const int n0 = wave * 16, nc = n0 + lo;
  const int b0 = blockIdx.x * BROWS;

  v2f bfr[64];  // fragments of M (already in x_new = x @ M layout)
#pragma unroll
  for (int kb = 0; kb < 64; ++kb) {
    const int k0 = 4 * kb + 2 * hi;
    bfr[kb].x = M[(size_t)k0 * Nn + nc];
    bfr[kb].y = M[(size_t)(k0 + 1) * Nn + nc];
  }

  // B_0 = S_0
  for (int i = tid; i < BROWS * Nn; i += 512) {
    int r = i >> 8, c = i & 255;
    float v = out[((size_t)(CHUNK - 1) * Bb + b0 + r) * Nn + c];
    xb[0][r * XSTR + c] = v;
    Bws[(size_t)(b0 + r) * Nn + c] = v;  // slot 0
  }
  __syncthreads();

  int cur = 0;
#pragma unroll 1
  for (int k = 1; k < NCHUNK; ++k) {
    const float* sk =
        out + ((size_t)(k * CHUNK + CHUNK - 1) * Bb + b0 + 8 * hi) * Nn + nc;
    v8f c0, c1 = zero8(), c2 = zero8(), c3 = zero8();
#pragma unroll
    for (int v = 0; v < 8; ++v) c0[v] = sk[(size_t)v * Nn];

    const float* xc = &xb[cur][lo * XSTR + 2 * hi];
#pragma unroll
    for (int kb = 0; kb < 64; kb += 4) {
      v2f a0 = *(const v2f*)(xc + 4 * kb);
      v2f a1 = *(const v2f*)(xc + 4 * kb + 4);
      v2f a2 = *(const v2f*)(xc + 4 * kb + 8);
      v2f a3 = *(const v2f*)(xc + 4 * kb + 12);
      c0 = wmma4(a0, bfr[kb + 0], c0);
      c1 = wmma4(a1, bfr[kb + 1], c1);
      c2 = wmma4(a2, bfr[kb + 2], c2);
      c3 = wmma4(a3, bfr[kb + 3], c3);
    }
    v8f c = (c0 + c1) + (c2 + c3);

    float* xn = xb[cur ^ 1];
#pragma unroll
    for (int v = 0; v < 8; ++v) {
      xn[(v + 8 * hi) * XSTR + nc] = c[v];
      Bws[((size_t)k * Bb + b0 + v + 8 * hi) * Nn + nc] = c[v];
    }
    __syncthreads();
    cur ^= 1;
  }
}

// ---------------------------------------------------------------- phase 3 ---
// For chunk k >= 1: x = B_{k-1}; repeat x = x @ dA^T, out[t] += x.
__global__ __launch_bounds__(512, 1)
void k_correct(const float* __restrict__ dA, const float* __restrict__ Bws,
               float* __restrict__ out) {
  __shared__ float xb[2][BROWS * XSTR];
  const int tid = threadIdx.x, lane = tid & 31, wave = tid >> 5;
  const int lo = lane & 15, hi = lane >> 4;
  const int n0 = wave * 16, nc = n0 + lo;
  const int b0 = blockIdx.x * BROWS;
  const int k = blockIdx.y + 1;  // chunks 1..31
  const int t0 = k * CHUNK;

  v2f bfr[64];
#pragma unroll
  for (int kb = 0; kb < 64; ++kb) {
    const float* p = dA + (size_t)nc * Nn + 4 * kb + 2 * hi;
    bfr[kb].x = p[0];
    bfr[kb].y = p[1];
  }

  for (int i = tid; i < BROWS * Nn; i += 512) {
    int r = i >> 8, c = i & 255;
    xb[0][r * XSTR + c] = Bws[((size_t)(k - 1) * Bb + b0 + r) * Nn + c];
  }
  __syncthreads();

  int cur = 0;
#pragma unroll 1
  for (int j = 0; j < CHUNK; ++j) {
    v8f c0 = zero8(), c1 = zero8(), c2 = zero8(), c3 = zero8();
    const float* xc = &xb[cur][lo * XSTR + 2 * hi];
#pragma unroll
    for (int kb = 0; kb < 64; kb += 4) {
      v2f a0 = *(const v2f*)(xc + 4 * kb);
      v2f a1 = *(const v2f*)(xc + 4 * kb + 4);
      v2f a2 = *(const v2f*)(xc + 4 * kb + 8);
      v2f a3 = *(const v2f*)(xc + 4 * kb + 12);
      c0 = wmma4(a0, bfr[kb + 0], c0);
      c1 = wmma4(a1, bfr[kb + 1], c1);
      c2 = wmma4(a2, bfr[kb + 2], c2);
      c3 = wmma4(a3, bfr[kb + 3], c3);
    }
    v8f c = (c0 + c1) + (c2 + c3);

    float* xn = xb[cur ^ 1];
    float* og = out + ((size_t)(t0 + j) * Bb + b0 + 8 * hi) * Nn + nc;
#pragma unroll
    for (int v = 0; v < 8; ++v) {
      xn[(v + 8 * hi) * XSTR + nc] = c[v];
      og[(size_t)v * Nn] += c[v];
    }
    __syncthreads();
    cur ^= 1;
  }
}

// ------------------------------------------------------------------ launch --
extern "C" void kernel_launch(void* const* d_in, const int* in_sizes, int n_in,
                              void* d_out, int out_size, void* d_ws,
                              size_t ws_size, hipStream_t stream) {
  const float* inp = (const float*)d_in[0];  // (L, BATCH)
  const float* dA = (const float*)d_in[1];   // (N, N)
  const float* dB = (const float*)d_in[2];   // (N,)
  float* out = (float*)d_out;                // (L, BATCH, N)

  // Workspace layout (floats): P0[256*256] | P1[256*256] | Bws[32*128*256]
  float* P0 = (float*)d_ws;
  float* P1 = P0 + Nn * Nn;
  float* Bws = P1 + Nn * Nn;

  // Phase 0: M = (dA^T)^64 by repeated squaring -> ends in P0
  k_transpose<<<256, 256, 0, stream>>>(dA, P0);
  k_matmul256<<<16, 512, 0, stream>>>(P0, P0, P1);  // G^2
  k_matmul256<<<16, 512, 0, stream>>>(P1, P1, P0);  // G^4
  k_matmul256<<<16, 512, 0, stream>>>(P0, P0, P1);  // G^8
  k_matmul256<<<16, 512, 0, stream>>>(P1, P1, P0);  // G^16
  k_matmul256<<<16, 512, 0, stream>>>(P0, P0, P1);  // G^32
  k_matmul256<<<16, 512, 0, stream>>>(P1, P1, P0);  // G^64

  // Phase 1: parallel chunk-local scans (256 workgroups)
  k_scan_local<<<dim3(NBB, NCHUNK), 512, 0, stream>>>(inp, dA, dB, out);

  // Phase 2: chunk-boundary states (sequential in k, parallel over batch)
  k_boundary<<<NBB, 512, 0, stream>>>(out, P0, Bws);

  // Phase 3: parallel corrections (248 workgroups)
  k_correct<<<dim3(NBB, NCHUNK - 1), 512, 0, stream>>>(dA, Bws, out);
}